// QuantumLayer_70342974374007
// MI455X (gfx1250) — compile-verified
//
#include <hip/hip_runtime.h>
#include <hip/hip_bf16.h>

typedef float v2f __attribute__((ext_vector_type(2)));
typedef float v8f __attribute__((ext_vector_type(8)));

// ---------------------------------------------------------------------------
// Kernel 1: build U = (Perm_6 R_6) ... (Perm_1 R_1) from weights (6x4x3).
// 16 threads, thread j owns column j of U (gates act row-wise -> fully local).
// Writes U_r / U_i directly in WMMA 16x16x4 A-fragment order:
//   frag element for U[m][jcol]:  t=jcol>>2, lane = m + ((jcol&2)?16:0), comp=jcol&1
//   flat float index: (t*32 + lane)*2 + comp   (U_r at ws[0..255], U_i at ws[256..511])
// ---------------------------------------------------------------------------
__global__ void build_u_kernel(const float* __restrict__ w, float* __restrict__ ws) {
    const int j = threadIdx.x;
    if (j >= 16) return;
    float cr[16], ci[16];
    for (int k = 0; k < 16; ++k) { cr[k] = (k == j) ? 1.f : 0.f; ci[k] = 0.f; }

    for (int l = 0; l < 6; ++l) {
        for (int i = 0; i < 4; ++i) {
            const float phi = w[(l * 4 + i) * 3 + 0];
            const float th  = w[(l * 4 + i) * 3 + 1];
            const float om  = w[(l * 4 + i) * 3 + 2];
            const float ct = cosf(0.5f * th), st = sinf(0.5f * th);
            const float a  = 0.5f * (phi + om), bb = 0.5f * (phi - om);
            const float ca = cosf(a), sa = sinf(a), cb = cosf(bb), sb = sinf(bb);
            // Rot gate: g00=ct e^{-ia}  g01=-st e^{+ib}  g10=st e^{-ib}  g11=ct e^{+ia}
            const float g00r =  ct * ca, g00i = -ct * sa;
            const float g01r = -st * cb, g01i = -st * sb;
            const float g10r =  st * cb, g10i = -st * sb;
            const float g11r =  ct * ca, g11i =  ct * sa;
            const int bit = 1 << (3 - i);
            for (int j0 = 0; j0 < 16; ++j0) {
                if (j0 & bit) continue;
                const int j1 = j0 | bit;
                const float v0r = cr[j0], v0i = ci[j0], v1r = cr[j1], v1i = ci[j1];
                cr[j0] = g00r * v0r - g00i * v0i + g01r * v1r - g01i * v1i;
                ci[j0] = g00r * v0i + g00i * v0r + g01r * v1i + g01i * v1r;
                cr[j1] = g10r * v0r - g10i * v0i + g11r * v1r - g11i * v1i;
                ci[j1] = g10r * v0i + g10i * v0r + g11r * v1i + g11i * v1r;
            }
        }
        // ring permutation for this layer: rows k <- ring[k]
        const int r = (l % 3) + 1;
        int perm[16], np[16];
        for (int k = 0; k < 16; ++k) perm[k] = k;
        for (int i = 0; i < 4; ++i) {
            const int tgt = (i + r) & 3;
            for (int k = 0; k < 16; ++k) {
                const int cbit = (k >> (3 - i)) & 1;
                const int src  = cbit ? (k ^ (1 << (3 - tgt))) : k;
                np[k] = perm[src];
            }
            for (int k = 0; k < 16; ++k) perm[k] = np[k];
        }
        float tr[16], ti[16];
        for (int k = 0; k < 16; ++k) { tr[k] = cr[perm[k]]; ti[k] = ci[perm[k]]; }
        for (int k = 0; k < 16; ++k) { cr[k] = tr[k]; ci[k] = ti[k]; }
    }

    // scatter column into A-fragment layout
    const int t     = j >> 2;
    const int lbase = (j & 2) ? 16 : 0;
    const int comp  = j & 1;
    float* wr = ws;
    float* wi = ws + 256;
    for (int m = 0; m < 16; ++m) {
        const int L = lbase + m;
        wr[(t * 32 + L) * 2 + comp] = cr[m];
        wi[(t * 32 + L) * 2 + comp] = ci[m];
    }
}

// ---------------------------------------------------------------------------
// Kernel 2: one wave32 per 16-batch tile.
//   S_r/S_i (16 states x 16 batch) = U_r/U_i x E   via 8x V_WMMA_F32_16X16X4_F32
//   then probs, signed Z reductions, shfl_xor(16) combine, float4 store.
// ---------------------------------------------------------------------------
__global__ void __launch_bounds__(256)
qlayer_kernel(const float* __restrict__ x, const float* __restrict__ ws,
              float* __restrict__ out, int batch) {
    const int lane = threadIdx.x & 31;
    const int wave = (blockIdx.x * blockDim.x + threadIdx.x) >> 5;
    int b = wave * 16 + (lane & 15);
    if (b >= batch) b = batch - 1;   // clamp (batch is a multiple of 16 anyway)

    // per-element embedding factors
    const float4 xv = *reinterpret_cast<const float4*>(x + (size_t)b * 4);
    const float c0 = __cosf(0.5f * xv.x), s0 = __sinf(0.5f * xv.x);
    const float c1 = __cosf(0.5f * xv.y), s1 = __sinf(0.5f * xv.y);
    const float c2 = __cosf(0.5f * xv.z), s2 = __sinf(0.5f * xv.z);
    const float c3 = __cosf(0.5f * xv.w), s3 = __sinf(0.5f * xv.w);

    // B fragments (4x16 slices of E): lane<16 covers K=4t+{0,1}, lane>=16 K=4t+{2,3}
    const float hi = (lane < 16) ? c2 : s2;   // bit1 of j
    const float T0 = c0 * c1, T1 = c0 * s1, T2 = s0 * c1, T3 = s0 * s1; // bits3,2
    v2f bf[4];
    bf[0] = (v2f){T0 * hi * c3, T0 * hi * s3};
    bf[1] = (v2f){T1 * hi * c3, T1 * hi * s3};
    bf[2] = (v2f){T2 * hi * c3, T2 * hi * s3};
    bf[3] = (v2f){T3 * hi * c3, T3 * hi * s3};

    // A fragments (pre-shuffled by build_u_kernel) — coalesced b64 loads
    const float* wr = ws;
    const float* wi = ws + 256;
    v2f ar[4], ai[4];
#pragma unroll
    for (int t = 0; t < 4; ++t) {
        ar[t] = *reinterpret_cast<const v2f*>(wr + (t * 32 + lane) * 2);
        ai[t] = *reinterpret_cast<const v2f*>(wi + (t * 32 + lane) * 2);
    }

    v8f accr = {0.f, 0.f, 0.f, 0.f, 0.f, 0.f, 0.f, 0.f};
    v8f acci = {0.f, 0.f, 0.f, 0.f, 0.f, 0.f, 0.f, 0.f};
#pragma unroll
    for (int t = 0; t < 4; ++t) {
        accr = __builtin_amdgcn_wmma_f32_16x16x4_f32(
            false, ar[t], false, bf[t], (short)0, accr, false, false);
        acci = __builtin_amdgcn_wmma_f32_16x16x4_f32(
            false, ai[t], false, bf[t], (short)0, acci, false, false);
    }

    // probabilities and signed reductions over this lane's 8 rows
    const int mbase = (lane < 16) ? 0 : 8;
    float o0 = 0.f, o1 = 0.f, o2 = 0.f, o3 = 0.f;
#pragma unroll
    for (int v = 0; v < 8; ++v) {
        const float p = accr[v] * accr[v] + acci[v] * acci[v];
        const int m = mbase + v;
        o0 += ((m >> 3) & 1) ? -p : p;
        o1 += ((m >> 2) & 1) ? -p : p;
        o2 += ((m >> 1) & 1) ? -p : p;
        o3 += ( m       & 1) ? -p : p;
    }
    // combine rows 0-7 (lane) with rows 8-15 (lane^16)
    o0 += __shfl_xor(o0, 16, 32);
    o1 += __shfl_xor(o1, 16, 32);
    o2 += __shfl_xor(o2, 16, 32);
    o3 += __shfl_xor(o3, 16, 32);

    if (lane < 16 && b < batch) {
        float4 res = {o0, o1, o2, o3};
        *reinterpret_cast<float4*>(out + (size_t)b * 4) = res;
    }
}

extern "C" void kernel_launch(void* const* d_in, const int* in_sizes, int n_in,
                              void* d_out, int out_size, void* d_ws, size_t ws_size,
                              hipStream_t stream) {
    const float* x = (const float*)d_in[0];        // (BATCH, 4) f32
    const float* w = (const float*)d_in[1];        // (6, 4, 3)  f32
    float* out = (float*)d_out;                    // (BATCH, 4) f32
    float* ws  = (float*)d_ws;                     // 512 floats used

    const int batch = in_sizes[0] / 4;

    build_u_kernel<<<1, 16, 0, stream>>>(w, ws);

    const int waves  = (batch + 15) / 16;          // one wave per 16 elements
    const int wpb    = 8;                          // 256 threads/block
    const int blocks = (waves + wpb - 1) / wpb;
    qlayer_kernel<<<blocks, 256, 0, stream>>>(x, ws, out, batch);
}